// VectorQuantizer_54271206752615
// MI455X (gfx1250) — compile-verified
//
#include <hip/hip_runtime.h>
#include <stdint.h>

typedef __attribute__((ext_vector_type(2))) float v2f;
typedef __attribute__((ext_vector_type(8))) float v8f;

#define D_DIM     512
#define K_CODES   16
#define KCHUNK    64
#define NCHUNKS   (D_DIM / KCHUNK)          // 8
#define ROWS_BLK  64
#define ROWS_WAVE 16
#define NTHREADS  128
#define ZSTRIDE   68                        // 68 % 64 == 4 -> conflict-free fragment reads

__global__ void vq_zero_loss_kernel(float* p) {
    if (threadIdx.x == 0) *p = 0.0f;
}

// flat pointer to LDS: low 32 bits are the wave-relative LDS byte offset
__device__ __forceinline__ uint32_t lds_off(const void* p) {
    return (uint32_t)(uintptr_t)p;
}
// CDNA5 async copy: ASYNCcnt-tracked, bypasses VGPRs (GV addressing mode)
__device__ __forceinline__ void async_load_b128(uint32_t lds, const float* g) {
    asm volatile("global_load_async_to_lds_b128 %0, %1, off"
                 :: "v"(lds), "v"((uint64_t)(uintptr_t)g) : "memory");
}
__device__ __forceinline__ void wait_async0() {
    asm volatile("s_wait_asynccnt 0x0" ::: "memory");
}

__global__ __launch_bounds__(NTHREADS)
void vq_wmma_kernel(const float* __restrict__ z,
                    const float* __restrict__ codebook,
                    float* __restrict__ zq,
                    float* __restrict__ idxOut,
                    float* __restrict__ lossOut,
                    float lossScale)
{
    __shared__ __align__(16) float sZ[2][ROWS_BLK][ZSTRIDE];   // double-buffered z chunks
    __shared__ __align__(16) float sC[2][K_CODES][ZSTRIDE];    // double-buffered codebook chunks
    __shared__ int   sIdx[ROWS_BLK];
    __shared__ float sLoss;

    const int tid  = threadIdx.x;
    const int wave = tid >> 5;
    const int lane = tid & 31;
    const int n    = lane & 15;             // fragment row (A) / column (B,C)
    const int half = lane >> 4;             // K-slice selector within the wave
    const int blockRow0 = blockIdx.x * ROWS_BLK;

    if (tid == 0) sLoss = 0.0f;

    auto issue_chunk = [&](int kc, int buf) {
        const float* zBlk = z + (size_t)blockRow0 * D_DIM + kc * KCHUNK;
        #pragma unroll
        for (int it = 0; it < 8; ++it) {
            int idx = it * NTHREADS + tid;  // 0..1023 float4 slots (64 rows x 16)
            int r = idx >> 4, c4 = idx & 15;
            async_load_b128(lds_off(&sZ[buf][r][c4 * 4]),
                            zBlk + (size_t)r * D_DIM + c4 * 4);
        }
        const float* cBlk = codebook + kc * KCHUNK;
        #pragma unroll
        for (int it = 0; it < 2; ++it) {
            int idx = it * NTHREADS + tid;  // 0..255 float4 slots (16 rows x 16)
            int r = idx >> 4, c4 = idx & 15;
            async_load_b128(lds_off(&sC[buf][r][c4 * 4]),
                            cBlk + (size_t)r * D_DIM + c4 * 4);
        }
    };

    // prologue: stage chunk 0
    issue_chunk(0, 0);
    wait_async0();
    __syncthreads();

    v8f acc = {0.f, 0.f, 0.f, 0.f, 0.f, 0.f, 0.f, 0.f};
    v2f znv = {0.f, 0.f};                   // |z_row|^2 partials, kept packed -> v_pk_fma_f32
    v2f cnv = {0.f, 0.f};                   // |c_n|^2 partials

    for (int kc = 0; kc < NCHUNKS; ++kc) {
        const int cur = kc & 1;
        if (kc + 1 < NCHUNKS) issue_chunk(kc + 1, cur ^ 1);   // overlap with WMMA below

        // A layout (16x4 f32): lane L -> row L&15; VGPR0 K=2*(L>>4), VGPR1 K+1. B symmetric.
        const float* aRow = &sZ[cur][wave * ROWS_WAVE + n][0];
        const float* bRow = &sC[cur][n][0];
        #pragma unroll
        for (int kk = 0; kk < KCHUNK / 4; ++kk) {
            const int kb = kk * 4 + half * 2;
            v2f a = *(const v2f*)(aRow + kb);
            v2f b = *(const v2f*)(bRow + kb);
            acc = __builtin_amdgcn_wmma_f32_16x16x4_f32(
                false, a, false, b, (short)0, acc, false, false);
            znv += a * a;                   // packed FMA on the fragment pair, co-executes with XDL
            cnv += b * b;
        }
        wait_async0();                      // own async loads for chunk kc+1 landed
        __syncthreads();                    // everyone's landed; done reading buf cur
    }

    // horizontal + cross-half: lane L and L^16 share n and cover complementary K slices
    float zn = znv.x + znv.y;
    float cn = cnv.x + cnv.y;
    const float znF = zn + __shfl_xor(zn, 16, 32);   // |z_{row n}|^2
    const float cnF = cn + __shfl_xor(cn, 16, 32);   // |c_n|^2

    // argmin over 16 codes; min distance == |z|^2 - 2 z.c + |c|^2 == loss contribution
    #pragma unroll
    for (int i = 0; i < 8; ++i) {
        const int m = half * 8 + i;         // row of acc[i] in the wave tile
        float val = cnF - 2.0f * acc[i];
        int   bi  = n;
        #pragma unroll
        for (int off = 8; off >= 1; off >>= 1) {     // stays inside each 16-lane half
            float ov = __shfl_xor(val, off, 32);
            int   oi = __shfl_xor(bi,  off, 32);
            if (ov < val || (ov == val && oi < bi)) { val = ov; bi = oi; }
        }
        if (n == m) {                       // lane that owns row m's norm
            sIdx[wave * ROWS_WAVE + m] = bi;
            atomicAdd(&sLoss, val + znF);
        }
    }
    __syncthreads();

    // ---- outputs ----
    if (tid == 0) atomicAdd(lossOut, sLoss * lossScale);
    if (tid < ROWS_BLK) idxOut[blockRow0 + tid] = (float)sIdx[tid];

    // z_q = codebook[idx]; 32KB codebook stays cache-hot, writes fully coalesced
    #pragma unroll 1
    for (int r = 0; r < ROWS_WAVE; ++r) {
        const int lr = wave * ROWS_WAVE + r;
        const float4* crow = (const float4*)(codebook + (size_t)sIdx[lr] * D_DIM);
        float4*       orow = (float4*)(zq + (size_t)(blockRow0 + lr) * D_DIM);
        #pragma unroll
        for (int it = 0; it < 4; ++it)
            orow[it * 32 + lane] = crow[it * 32 + lane];
    }
}

extern "C" void kernel_launch(void* const* d_in, const int* in_sizes, int n_in,
                              void* d_out, int out_size, void* d_ws, size_t ws_size,
                              hipStream_t stream) {
    const float* z  = (const float*)d_in[0];
    const float* cb = (const float*)d_in[1];

    const size_t zElems = (size_t)in_sizes[0];      // N * 512
    const int    nRows  = (int)(zElems / D_DIM);    // 65536

    float* zq      = (float*)d_out;
    float* idxOut  = zq + zElems;
    float* lossOut = idxOut + nRows;

    const float lossScale = 1.25f / ((float)nRows * (float)D_DIM);

    vq_zero_loss_kernel<<<1, 32, 0, stream>>>(lossOut);
    vq_wmma_kernel<<<nRows / ROWS_BLK, NTHREADS, 0, stream>>>(
        z, cb, zq, idxOut, lossOut, lossScale);
}